// final_block_nodc_81724637708525
// MI455X (gfx1250) — compile-verified
//
#include <hip/hip_runtime.h>

typedef __attribute__((ext_vector_type(2))) float v2f;
typedef __attribute__((ext_vector_type(8))) float v8f;

// db2 reconstruction taps (fp32 of numpy-derived values)
#define RL0  0.48296291314453414f
#define RL1  0.83651630373780790f
#define RL2  0.22414386804201338f
#define RL3 -0.12940952255126034f
#define RH0 -0.12940952255126034f
#define RH1 -0.22414386804201338f
#define RH2  0.83651630373780790f
#define RH3 -0.48296291314453414f

// Banded synthesis matrix W[16x18]: row i (pair s=i>>1, parity p=i&1) has
// nonzeros at j=2s..2s+3 with values {RL2,RH2,RL0,RH0} (p=0) / {RL3,RH3,RL1,RH1} (p=1).
__device__ __forceinline__ float wval(int i, int j) {
  int dj = j - (i & ~1);
  int p  = i & 1;
  float rl = (dj < 2) ? (p ? RL3 : RL2) : (p ? RL1 : RL0);
  float rh = (dj < 2) ? (p ? RH3 : RH2) : (p ? RH1 : RH0);
  float v  = (dj & 1) ? rh : rl;
  return ((unsigned)dj <= 3u) ? v : 0.0f;
}

__global__ void __launch_bounds__(256) copy_f4(float4* __restrict__ dst,
                                               const float4* __restrict__ src, int n4) {
  int i = blockIdx.x * blockDim.x + threadIdx.x;
  if (i < n4) dst[i] = src[i];
}

// Pass A: vertical (axis-0) inverse DWT via V_WMMA_F32_16X16X4_F32.
// One wave computes one 16x16 tile of L (from LL,HL) or H (from LH,HH):
//   D(16x16) = sum_{c=0..4} Wc(16x4) x Gc(4x16)
// G rows j: even -> lo[(t0-1+j/2) mod m], odd -> hi[(t0-1+j/2) mod m], cols j0..j0+15.
__global__ void __launch_bounds__(256) idwt_pass_col(
    const float* __restrict__ buf, float* __restrict__ Lws, float* __restrict__ Hws,
    int m, int n, int log2TilesN, int totalTiles)
{
  const int S = 4096;                  // row stride of the wt buffer
  int wave = threadIdx.x >> 5;
  int lane = threadIdx.x & 31;
  int tileIdx = blockIdx.x * 8 + wave; // wave-uniform
  if (tileIdx >= totalTiles) return;   // uniform per wave -> EXEC stays all-1s

  int perArray = totalTiles >> 1;
  int isH  = tileIdx >= perArray;
  int tIdx = isH ? (tileIdx - perArray) : tileIdx;
  int tileM = tIdx >> log2TilesN;
  int tileN = tIdx & ((1 << log2TilesN) - 1);

  const float* loB = buf + (isH ? n : 0);        // LL or LH quadrant
  const float* hiB = loB + (size_t)m * S;        // HL or HH quadrant
  float*       outB = isH ? Hws : Lws;

  int t0    = tileM << 3;              // first output pair index of tile
  int col   = (tileN << 4) + (lane & 15);
  int khalf = lane >> 4;               // lane group selects K pair within chunk
  int iRow  = lane & 15;               // A-matrix M index for this lane

  v8f acc = {};
#pragma unroll
  for (int c = 0; c < 5; ++c) {
    // gathered input rows for this lane's K slice (same s for both K's)
    int s   = 2 * c + khalf;
    int row = t0 - 1 + s;
    row = (row < 0) ? (row + m) : ((row >= m) ? (row - m) : row);
    v2f B;
    B.x = loB[(size_t)row * S + col];  // j = 4c+2*khalf   (even -> lo)
    B.y = hiB[(size_t)row * S + col];  // j = 4c+2*khalf+1 (odd  -> hi)
    // constant A slice (zero for padded j>=18, which nulls the extra B loads)
    int jA = 4 * c + 2 * khalf;
    v2f A;
    A.x = wval(iRow, jA);
    A.y = wval(iRow, jA + 1);
    acc = __builtin_amdgcn_wmma_f32_16x16x4_f32(false, A, false, B,
                                                (short)0, acc, false, false);
  }

  // C/D layout: VGPR r holds M = r + 8*khalf, N = lane&15
  size_t base = (size_t)(tileM << 4) * n + col;
#pragma unroll
  for (int r = 0; r < 8; ++r)
    outB[base + (size_t)(r + 8 * khalf) * n] = acc[r];
}

// Pass B: horizontal (axis-1) inverse DWT, VALU, one thread per output pair.
__global__ void __launch_bounds__(256) idwt_pass_row(
    float* __restrict__ buf, const float* __restrict__ Lws, const float* __restrict__ Hws,
    int n, int log2n, int total)
{
  int gid = blockIdx.x * 256 + threadIdx.x;
  if (gid >= total) return;
  int u = gid & (n - 1);
  int i = gid >> log2n;
  int um1 = (u == 0) ? (n - 1) : (u - 1);
  size_t rb = (size_t)i * n;
  float l  = Lws[rb + u], lm = Lws[rb + um1];
  float h  = Hws[rb + u], hm = Hws[rb + um1];
  float2 o;
  o.x = RL0 * l + RL2 * lm + RH0 * h + RH2 * hm;
  o.y = RL1 * l + RL3 * lm + RH1 * h + RH3 * hm;
  *reinterpret_cast<float2*>(buf + (size_t)i * 4096 + 2 * u) = o;
}

extern "C" void kernel_launch(void* const* d_in, const int* in_sizes, int n_in,
                              void* d_out, int out_size, void* d_ws, size_t ws_size,
                              hipStream_t stream) {
  (void)in_sizes; (void)n_in; (void)out_size; (void)ws_size;
  const float* x = (const float*)d_in[0];       // (1,4096,4096) fp32
  float* out = (float*)d_out;                   // evolving wt buffer (4096x4096)
  float* ws  = (float*)d_ws;                    // L/H intermediates (<= 64 MB)
  const int W = 4096;

  int n4 = (W * W) / 4;
  copy_f4<<<(n4 + 255) / 256, 256, 0, stream>>>((float4*)out, (const float4*)x, n4);

  for (int lev = 4; lev >= 1; --lev) {
    int m = W >> lev;            // quadrant height
    int n = m;                   // quadrant width
    int log2n = 12 - lev;
    float* Lws = ws;
    float* Hws = ws + (size_t)2 * m * n;

    int totalTiles = (m * n) >> 6;        // 2 * (2m/16) * (n/16)
    int log2TilesN = log2n - 4;
    idwt_pass_col<<<(totalTiles + 7) / 8, 256, 0, stream>>>(
        out, Lws, Hws, m, n, log2TilesN, totalTiles);

    int total = 2 * m * n;                // one thread per (row, u-pair)
    idwt_pass_row<<<(total + 255) / 256, 256, 0, stream>>>(
        out, Lws, Hws, n, log2n, total);
  }
}